// IRMC_NN_Model_80290118631949
// MI455X (gfx1250) — compile-verified
//
#include <hip/hip_runtime.h>
#include <hip/hip_bf16.h>
#include <math.h>

typedef __attribute__((ext_vector_type(2))) float v2f;
typedef __attribute__((ext_vector_type(8))) float v8f;

// ---- problem constants ----
constexpr int N_USER = 1000000;
constexpr int N_ITEM = 100000;
constexpr int E      = 16;
constexpr int O      = 16;
constexpr int H      = 4;
constexpr int HID    = 32;
constexpr int B      = 8192;
constexpr int S      = 200;
constexpr int L      = 50;
constexpr int NSUPP  = 50000;

// ---- workspace layout (floats) ----
constexpr size_t O_UI  = 0;                          // user_init   B*16
constexpr size_t O_K   = O_UI  + (size_t)B*16;       // K_supp      H*NSUPP*16
constexpr size_t O_Q   = O_K   + (size_t)H*NSUPP*16; // Q           H*B*16
constexpr size_t O_CTX = O_Q   + (size_t)H*B*16;     // ctx         H*B*16
constexpr size_t O_M   = O_CTX + (size_t)H*B*16;     // M           H*16*16
constexpr size_t O_UE  = O_M   + (size_t)H*16*16;    // user_emb    B*16

// =====================================================================
// Exact f32 16x16x16 GEMM tile via four chained V_WMMA_F32_16X16X4_F32.
// A (16x4 f32) layout: lanes 0-15 -> row m=lane, VGPR0 K=0, VGPR1 K=1;
//                      lanes 16-31 -> row m=lane-16, VGPR0 K=2, VGPR1 K=3.
// B (4x16) mirrored per column n = lane&15 with the same K mapping.
// =====================================================================
__device__ inline v8f wmma_f32_tile(const float As[16][16],
                                    const float Bs[16][16],
                                    v8f c, int lane) {
  const int m    = lane & 15;
  const int koff = (lane < 16) ? 0 : 2;
#pragma unroll
  for (int kc = 0; kc < 4; ++kc) {
    v2f a, b;
    a[0] = As[m][kc * 4 + koff];
    a[1] = As[m][kc * 4 + koff + 1];
    b[0] = Bs[kc * 4 + koff][m];
    b[1] = Bs[kc * 4 + koff + 1][m];
    c = __builtin_amdgcn_wmma_f32_16x16x4_f32(
        /*neg_a=*/false, a, /*neg_b=*/false, b,
        /*c_mod=*/(short)0, c, /*reuse_a=*/false, /*reuse_b=*/false);
  }
  return c;
}

// C/D f32 16x16 layout: VGPR v -> row (v + (lane<16?0:8)), col = lane&15.
// Bounce through LDS so the global store is 2 coalesced float4 per lane.
__device__ inline void store_tile_coalesced(float* __restrict__ dst, v8f c,
                                            int lane, float Ds[16][16]) {
  const int n  = lane & 15;
  const int mb = (lane < 16) ? 0 : 8;
#pragma unroll
  for (int v = 0; v < 8; ++v) Ds[mb + v][n] = c[v];
  __syncthreads();
  const float4* src = (const float4*)&Ds[0][0];
  float4* d4 = (float4*)dst;
  d4[lane]      = src[lane];
  d4[lane + 32] = src[lane + 32];
}

// =====================================================================
// K1: user_init[b,e] = (sum_l item_emb[history[b,l], e]) / history_len[b]
// =====================================================================
__global__ void user_init_kernel(const float* __restrict__ item_tab,
                                 const int* __restrict__ history,
                                 const int* __restrict__ hist_len,
                                 float* __restrict__ ui) {
  int g = blockIdx.x * 256 + threadIdx.x;  // B*16 threads
  int b = g >> 4, e = g & 15;
  const int* hrow = history + (size_t)b * L;
  float s = 0.f;
#pragma unroll 5
  for (int l = 0; l < L; ++l) s += item_tab[(size_t)hrow[l] * E + e];
  ui[g] = s / (float)hist_len[b];
}

// =====================================================================
// K2: K_supp[h, n, :] = user_emb_tab[supp_users[n]] @ wk[h]   (f32 WMMA)
// grid = (NSUPP/16, H), block = 32
// =====================================================================
__global__ void ksupp_kernel(const float* __restrict__ utab,
                             const int* __restrict__ supp,
                             const float* __restrict__ wk,
                             float* __restrict__ K) {
  __shared__ float As[16][16];
  __shared__ float Bs[16][16];
  __shared__ float Ds[16][16];
  const int tile = blockIdx.x, h = blockIdx.y, lane = threadIdx.x;
#pragma unroll
  for (int i = lane; i < 256; i += 32) {
    int r = i >> 4, cdx = i & 15;
    As[r][cdx] = utab[(size_t)supp[tile * 16 + r] * E + cdx];
    Bs[r][cdx] = wk[(size_t)h * 256 + i];  // wk[h][e][o] -> Bs[k][n]
  }
  __syncthreads();
  v8f c = {};
  c = wmma_f32_tile(As, Bs, c, lane);
  store_tile_coalesced(K + ((size_t)h * NSUPP + (size_t)tile * 16) * 16, c, lane, Ds);
}

// =====================================================================
// K3: Q[h, b, :] = user_init[b] @ wq[h]   (f32 WMMA)
// grid = (B/16, H), block = 32
// =====================================================================
__global__ void q_kernel(const float* __restrict__ ui,
                         const float* __restrict__ wq,
                         float* __restrict__ Q) {
  __shared__ float As[16][16];
  __shared__ float Bs[16][16];
  __shared__ float Ds[16][16];
  const int tile = blockIdx.x, h = blockIdx.y, lane = threadIdx.x;
#pragma unroll
  for (int i = lane; i < 256; i += 32) {
    int r = i >> 4, cdx = i & 15;
    As[r][cdx] = ui[(size_t)(tile * 16 + r) * 16 + cdx];
    Bs[r][cdx] = wq[(size_t)h * 256 + i];
  }
  __syncthreads();
  v8f c = {};
  c = wmma_f32_tile(As, Bs, c, lane);
  store_tile_coalesced(Q + ((size_t)h * B + (size_t)tile * 16) * 16, c, lane, Ds);
}

// =====================================================================
// K4: attention — one wave32 per (h,b). Lane-per-sample, S=200 in 7 strides.
//   logits -> wave-max -> exp/sum -> weighted K accumulation -> butterfly.
// =====================================================================
__global__ void attn_kernel(const float* __restrict__ Q,
                            const float* __restrict__ K,
                            const int* __restrict__ sidx,
                            float* __restrict__ ctx) {
  const int wave = threadIdx.x >> 5;
  const int lane = threadIdx.x & 31;
  const int p = blockIdx.x * 8 + wave;  // 0 .. H*B-1
  const int h = p >> 13;                // B = 2^13
  const int b = p & (B - 1);

  const float* q = Q + ((size_t)h * B + b) * 16;
  const float4 q0 = ((const float4*)q)[0];
  const float4 q1 = ((const float4*)q)[1];
  const float4 q2 = ((const float4*)q)[2];
  const float4 q3 = ((const float4*)q)[3];

  const int* si = sidx + ((size_t)h * B + b) * S;
  const float* Kh = K + (size_t)h * NSUPP * 16;

  float lg[7];
  float pmax = -3.4e38f;
#pragma unroll
  for (int it = 0; it < 7; ++it) {
    const int s = lane + 32 * it;
    float d = -3.4e38f;
    if (s < S) {
      if (it + 1 < 7) __builtin_prefetch(si + s + 32, 0, 0);  // global_prefetch_b8
      const int idx = si[s];
      const float4* kr = (const float4*)(Kh + (size_t)idx * 16);
      const float4 k0 = kr[0], k1 = kr[1], k2 = kr[2], k3 = kr[3];
      d = q0.x * k0.x + q0.y * k0.y + q0.z * k0.z + q0.w * k0.w +
          q1.x * k1.x + q1.y * k1.y + q1.z * k1.z + q1.w * k1.w +
          q2.x * k2.x + q2.y * k2.y + q2.z * k2.z + q2.w * k2.w +
          q3.x * k3.x + q3.y * k3.y + q3.z * k3.z + q3.w * k3.w;
    }
    lg[it] = d;
    pmax = fmaxf(pmax, d);
  }
#pragma unroll
  for (int off = 16; off > 0; off >>= 1)
    pmax = fmaxf(pmax, __shfl_xor(pmax, off, 32));

  float wsum = 0.f;
  float wgt[7];
#pragma unroll
  for (int it = 0; it < 7; ++it) {
    float e = (lg[it] > -1.0e37f) ? __expf(lg[it] - pmax) : 0.f;
    wgt[it] = e;
    wsum += e;
  }
#pragma unroll
  for (int off = 16; off > 0; off >>= 1) wsum += __shfl_xor(wsum, off, 32);
  const float inv = 1.f / wsum;

  float acc[16];
#pragma unroll
  for (int e = 0; e < 16; ++e) acc[e] = 0.f;
#pragma unroll
  for (int it = 0; it < 7; ++it) {
    const int s = lane + 32 * it;
    if (s < S) {
      const int idx = si[s];
      const float4* kr = (const float4*)(Kh + (size_t)idx * 16);
      const float4 k0 = kr[0], k1 = kr[1], k2 = kr[2], k3 = kr[3];
      const float w = wgt[it] * inv;
      acc[0] += w * k0.x;  acc[1] += w * k0.y;  acc[2] += w * k0.z;  acc[3] += w * k0.w;
      acc[4] += w * k1.x;  acc[5] += w * k1.y;  acc[6] += w * k1.z;  acc[7] += w * k1.w;
      acc[8] += w * k2.x;  acc[9] += w * k2.y;  acc[10] += w * k2.z; acc[11] += w * k2.w;
      acc[12] += w * k3.x; acc[13] += w * k3.y; acc[14] += w * k3.z; acc[15] += w * k3.w;
    }
  }
#pragma unroll
  for (int off = 16; off > 0; off >>= 1) {
#pragma unroll
    for (int e = 0; e < 16; ++e) acc[e] += __shfl_xor(acc[e], off, 32);
  }
  if (lane == 0) {
    float4* dst = (float4*)(ctx + ((size_t)h * B + b) * 16);
    dst[0] = make_float4(acc[0], acc[1], acc[2], acc[3]);
    dst[1] = make_float4(acc[4], acc[5], acc[6], acc[7]);
    dst[2] = make_float4(acc[8], acc[9], acc[10], acc[11]);
    dst[3] = make_float4(acc[12], acc[13], acc[14], acc[15]);
  }
}

// =====================================================================
// K5: M[h] = wv[h] @ w_out[h*16:(h+1)*16, :]   (tiny, 1024 outputs)
// =====================================================================
__global__ void m_kernel(const float* __restrict__ wv,
                         const float* __restrict__ wout,
                         float* __restrict__ M) {
  for (int i = threadIdx.x; i < H * 256; i += 256) {
    const int h = i >> 8, rem = i & 255, o = rem >> 4, pp = rem & 15;
    float acc = 0.f;
#pragma unroll
    for (int t = 0; t < 16; ++t)
      acc += wv[(size_t)h * 256 + o * 16 + t] * wout[(size_t)(h * 16 + t) * 16 + pp];
    M[i] = acc;
  }
}

// =====================================================================
// K6: user_emb[b] = sum_h ctx[h,b] @ M[h]   (4x4 chained f32 WMMAs)
// grid = B/16, block = 32
// =====================================================================
__global__ void uemb_kernel(const float* __restrict__ ctx,
                            const float* __restrict__ M,
                            float* __restrict__ ue) {
  __shared__ float As[16][16];
  __shared__ float Bs[16][16];
  __shared__ float Ds[16][16];
  const int tile = blockIdx.x, lane = threadIdx.x;
  v8f c = {};
#pragma unroll
  for (int h = 0; h < H; ++h) {
    __syncthreads();  // protect LDS reuse across iterations
#pragma unroll
    for (int i = lane; i < 256; i += 32) {
      int r = i >> 4, cdx = i & 15;
      As[r][cdx] = ctx[((size_t)h * B + tile * 16 + r) * 16 + cdx];
      Bs[r][cdx] = M[(size_t)h * 256 + i];
    }
    __syncthreads();
    c = wmma_f32_tile(As, Bs, c, lane);
  }
  __syncthreads();
  store_tile_coalesced(ue + (size_t)tile * 16 * 16, c, lane, Ds);
}

// =====================================================================
// K7: interaction + 48->32->16->1 tanh MLP + biases. One thread per b,
// MLP weights staged in LDS.
// =====================================================================
__global__ void final_kernel(const float* __restrict__ user_emb,
                             const float* __restrict__ item_tab,
                             const int* __restrict__ x,
                             const float* __restrict__ l1w, const float* __restrict__ l1b,
                             const float* __restrict__ l2w, const float* __restrict__ l2b,
                             const float* __restrict__ l3w, const float* __restrict__ l3b,
                             const float* __restrict__ ubias,
                             const float* __restrict__ ibias,
                             float* __restrict__ out) {
  __shared__ float s_l1w[48 * 32];
  __shared__ float s_l2w[32 * 16];
  __shared__ float s_l1b[32];
  __shared__ float s_l2b[16];
  __shared__ float s_l3w[16];
  const int tid = threadIdx.x;
  for (int i = tid; i < 48 * 32; i += 256) s_l1w[i] = l1w[i];
  for (int i = tid; i < 32 * 16; i += 256) s_l2w[i] = l2w[i];
  if (tid < 32) s_l1b[tid] = l1b[tid];
  if (tid < 16) { s_l2b[tid] = l2b[tid]; s_l3w[tid] = l3w[tid]; }
  __syncthreads();

  const int b = blockIdx.x * 256 + tid;
  const int uid = x[2 * b + 0];
  const int iid = x[2 * b + 1];

  float ue[16], ie[16], itx[16];
  float ratings = 0.f;
#pragma unroll
  for (int e = 0; e < 16; ++e) {
    ue[e] = user_emb[(size_t)b * 16 + e];
    ie[e] = item_tab[(size_t)iid * E + e];
    itx[e] = ue[e] * ie[e];
    ratings += itx[e];
  }
  float h1[32];
#pragma unroll
  for (int j = 0; j < 32; ++j) {
    float acc = s_l1b[j];
#pragma unroll
    for (int e = 0; e < 16; ++e) {
      acc += ue[e]  * s_l1w[e * 32 + j];
      acc += ie[e]  * s_l1w[(16 + e) * 32 + j];
      acc += itx[e] * s_l1w[(32 + e) * 32 + j];
    }
    h1[j] = tanhf(acc);
  }
  float x3 = l3b[0];
#pragma unroll
  for (int k = 0; k < 16; ++k) {
    float acc = s_l2b[k];
#pragma unroll
    for (int j = 0; j < 32; ++j) acc += h1[j] * s_l2w[j * 16 + k];
    x3 += tanhf(acc) * s_l3w[k];
  }
  out[b] = (ratings + x3) * 0.5f + ubias[uid] + ibias[iid];
}

// =====================================================================
extern "C" void kernel_launch(void* const* d_in, const int* in_sizes, int n_in,
                              void* d_out, int out_size, void* d_ws, size_t ws_size,
                              hipStream_t stream) {
  const float* user_embedding = (const float*)d_in[0];
  const float* item_embedding = (const float*)d_in[1];
  const float* wq    = (const float*)d_in[2];
  const float* wk    = (const float*)d_in[3];
  const float* wv    = (const float*)d_in[4];
  const float* w_out = (const float*)d_in[5];
  const float* l1_w  = (const float*)d_in[6];
  const float* l1_b  = (const float*)d_in[7];
  const float* l2_w  = (const float*)d_in[8];
  const float* l2_b  = (const float*)d_in[9];
  const float* l3_w  = (const float*)d_in[10];
  const float* l3_b  = (const float*)d_in[11];
  const float* user_bias = (const float*)d_in[12];
  const float* item_bias = (const float*)d_in[13];
  const int* x          = (const int*)d_in[14];
  const int* history    = (const int*)d_in[15];
  const int* hist_len   = (const int*)d_in[16];
  const int* supp_users = (const int*)d_in[17];
  const int* sample_idx = (const int*)d_in[18];

  float* W = (float*)d_ws;
  float* ui   = W + O_UI;
  float* K    = W + O_K;
  float* Q    = W + O_Q;
  float* ctx  = W + O_CTX;
  float* M    = W + O_M;
  float* uemb = W + O_UE;

  user_init_kernel<<<dim3(B * 16 / 256), 256, 0, stream>>>(item_embedding, history, hist_len, ui);
  ksupp_kernel<<<dim3(NSUPP / 16, H), 32, 0, stream>>>(user_embedding, supp_users, wk, K);
  q_kernel<<<dim3(B / 16, H), 32, 0, stream>>>(ui, wq, Q);
  attn_kernel<<<dim3(H * B / 8), 256, 0, stream>>>(Q, K, sample_idx, ctx);
  m_kernel<<<dim3(1), 256, 0, stream>>>(wv, w_out, M);
  uemb_kernel<<<dim3(B / 16), 32, 0, stream>>>(ctx, M, uemb);
  final_kernel<<<dim3(B / 256), 256, 0, stream>>>(uemb, item_embedding, x,
                                                  l1_w, l1_b, l2_w, l2_b, l3_w, l3_b,
                                                  user_bias, item_bias, (float*)d_out);
}